// GATTNET_Loss_23502061044108
// MI455X (gfx1250) — compile-verified
//
#include <hip/hip_runtime.h>
#include <hip/hip_bf16.h>
#include <math.h>

// Problem constants (from reference setup_inputs)
#define D_DIM 1024     // rows of H
#define N_DIM 8192     // cols of H
#define B_DIM 8192     // batch for CE
#define C_DIM 2        // classes
#define LAMBDA_COE 0.5f
#define EPS_NORM 1e-12f

typedef __attribute__((ext_vector_type(2))) float v2f;
typedef __attribute__((ext_vector_type(8))) float v8f;

// ---------------- workspace layout (floats) ----------------
// [0,            N_DIM)            : n2   per-column squared norms (accumulated)
// [N_DIM,        2*N_DIM)          : w    1/max(norm, eps)
// [2*N_DIM,      2*N_DIM+D_DIM)    : s    H @ w (accumulated)
// [2*N_DIM+D_DIM]                  : trace accumulator
// [2*N_DIM+D_DIM+1]                : ce-sum accumulator

__global__ void k_init(float* __restrict__ ws) {
    int i = blockIdx.x * blockDim.x + threadIdx.x;
    if (i < N_DIM) ws[i] = 0.0f;                         // n2
    if (i < D_DIM + 2) ws[2 * N_DIM + i] = 0.0f;         // s + accumulators
}

// Pass 1: per-column squared norms. 8 d-slabs for parallelism; coalesced along j.
__global__ void k_colnorm(const float* __restrict__ H, float* __restrict__ n2) {
    int j  = blockIdx.x * 256 + threadIdx.x;
    int d0 = blockIdx.y * (D_DIM / 8);
    const float* p = H + (size_t)d0 * N_DIM + j;
    float acc = 0.0f;
#pragma unroll 4
    for (int d = 0; d < D_DIM / 8; ++d) {
        float x = p[(size_t)d * N_DIM];
        acc = fmaf(x, x, acc);
    }
    atomicAdd(&n2[j], acc);
}

// Pass 1b: w_j = 1/max(||h_j||, eps); trace += ||h_j||^2 * w_j^2
__global__ void k_w(const float* __restrict__ n2, float* __restrict__ w,
                    float* __restrict__ trace_acc) {
    __shared__ float red[256];
    int j = blockIdx.x * 256 + threadIdx.x;
    float v   = n2[j];
    float nrm = fmaxf(sqrtf(v), EPS_NORM);
    float wi  = 1.0f / nrm;
    w[j] = wi;
    red[threadIdx.x] = v * wi * wi;
    __syncthreads();
    for (int o = 128; o > 0; o >>= 1) {
        if (threadIdx.x < o) red[threadIdx.x] += red[threadIdx.x + o];
        __syncthreads();
    }
    if (threadIdx.x == 0) atomicAdd(trace_acc, red[0]);
}

// Pass 2: s = H @ w via V_WMMA_F32_16X16X4_F32.
// One wave32 per block. blockIdx.x selects a 16-row d-tile, blockIdx.y a j-slab.
// A (16x4 f32):  lanes 0-15 hold M=0..15; a.x = K0 (lanes<16) / K2 (lanes>=16), a.y = K1/K3.
// B (4x16 f32):  broadcast w: VGPR0 = row K0 (lanes<16) / K2 (lanes>=16), VGPR1 = K1/K3.
// Every column of D equals the GEMV partial; extract from lanes 0 and 16.
__global__ __launch_bounds__(32) void k_gemv_wmma(const float* __restrict__ H,
                                                  const float* __restrict__ w,
                                                  float* __restrict__ s) {
    const int lane = threadIdx.x;                 // 0..31, wave32
    const int d0   = blockIdx.x * 16;
    const int jbeg = blockIdx.y * (N_DIM / 16);   // 512-column slab
    const int jend = jbeg + (N_DIM / 16);
    const int m    = lane & 15;                   // matrix row within tile
    const int kh   = (lane >> 4) << 1;            // 0 for lanes 0-15, 2 for lanes 16-31
    const float* rowp = H + (size_t)(d0 + m) * N_DIM;

    v8f c0 = {0.f, 0.f, 0.f, 0.f, 0.f, 0.f, 0.f, 0.f};
    v8f c1 = {0.f, 0.f, 0.f, 0.f, 0.f, 0.f, 0.f, 0.f};

    for (int j0 = jbeg; j0 < jend; j0 += 8) {
        v2f a0 = *(const v2f*)(rowp + j0 + kh);
        v2f b0 = *(const v2f*)(w    + j0 + kh);
        v2f a1 = *(const v2f*)(rowp + j0 + 4 + kh);
        v2f b1 = *(const v2f*)(w    + j0 + 4 + kh);
        // D = A x B + C, exact f32 accumulate (RNE). Two chains for ILP.
        c0 = __builtin_amdgcn_wmma_f32_16x16x4_f32(false, a0, false, b0,
                                                   (short)0, c0, false, false);
        c1 = __builtin_amdgcn_wmma_f32_16x16x4_f32(false, a1, false, b1,
                                                   (short)0, c1, false, false);
    }

    // C/D layout: lane n in [0,15]: VGPR r = D[M=r][N=n]; lane n in [16,31]: D[M=8+r][N=n-16].
    if (m == 0) {
        const int base = d0 + (lane >> 4) * 8;
#pragma unroll
        for (int r = 0; r < 8; ++r) atomicAdd(&s[base + r], c0[r] + c1[r]);
    }
}

// Cross-entropy over [B, 2] logits, mean reduction accumulated as a sum here.
__global__ void k_ce(const float* __restrict__ logits, const int* __restrict__ lab,
                     float* __restrict__ ce_acc) {
    __shared__ float red[256];
    int b = blockIdx.x * 256 + threadIdx.x;
    float x0 = logits[2 * b + 0];
    float x1 = logits[2 * b + 1];
    int   l  = lab[b];
    float mx  = fmaxf(x0, x1);
    float lse = mx + logf(expf(x0 - mx) + expf(x1 - mx));
    red[threadIdx.x] = lse - (l ? x1 : x0);
    __syncthreads();
    for (int o = 128; o > 0; o >>= 1) {
        if (threadIdx.x < o) red[threadIdx.x] += red[threadIdx.x + o];
        __syncthreads();
    }
    if (threadIdx.x == 0) atomicAdd(ce_acc, red[0]);
}

// Finalize: sum(G) = ||s||^2 ; loss = ce/B + (sumG - trace)/2 * lambda / (N(N-1)/2)
__global__ void k_final(const float* __restrict__ s, const float* __restrict__ acc,
                        float* __restrict__ out) {
    __shared__ float red[256];
    float a = 0.0f;
    for (int i = threadIdx.x; i < D_DIM; i += 256) {
        float v = s[i];
        a = fmaf(v, v, a);
    }
    red[threadIdx.x] = a;
    __syncthreads();
    for (int o = 128; o > 0; o >>= 1) {
        if (threadIdx.x < o) red[threadIdx.x] += red[threadIdx.x + o];
        __syncthreads();
    }
    if (threadIdx.x == 0) {
        float sumG     = red[0];
        float trace    = acc[0];
        float ce       = acc[1] / (float)B_DIM;
        float pair_sum = (sumG - trace) * 0.5f;
        float denom    = (float)N_DIM * (float)(N_DIM - 1) * 0.5f;
        out[0] = ce + pair_sum * LAMBDA_COE / denom;
    }
}

extern "C" void kernel_launch(void* const* d_in, const int* in_sizes, int n_in,
                              void* d_out, int out_size, void* d_ws, size_t ws_size,
                              hipStream_t stream) {
    const float* outputs = (const float*)d_in[0];   // [B, 2] f32
    const int*   labels  = (const int*)d_in[1];     // [B] int
    const float* H       = (const float*)d_in[2];   // [D, N] f32

    float* ws  = (float*)d_ws;
    float* n2  = ws;
    float* w   = ws + N_DIM;
    float* s   = ws + 2 * N_DIM;
    float* acc = ws + 2 * N_DIM + D_DIM;            // [0]=trace, [1]=ce-sum

    k_init<<<N_DIM / 256, 256, 0, stream>>>(ws);

    dim3 gcn(N_DIM / 256, 8);
    k_colnorm<<<gcn, 256, 0, stream>>>(H, n2);
    k_w<<<N_DIM / 256, 256, 0, stream>>>(n2, w, acc);

    k_ce<<<B_DIM / 256, 256, 0, stream>>>(outputs, labels, acc + 1);

    dim3 gg(D_DIM / 16, 16);                        // 64 d-tiles x 16 j-slabs = 1024 waves
    k_gemv_wmma<<<gg, 32, 0, stream>>>(H, w, s);

    k_final<<<1, 256, 0, stream>>>(s, acc, (float*)d_out);
}